// GraphNN_51625506898061
// MI455X (gfx1250) — compile-verified
//
#include <hip/hip_runtime.h>
#include <math.h>

typedef __attribute__((ext_vector_type(2))) float v2f;
typedef __attribute__((ext_vector_type(8))) float v8f;

// ---- monotone float <-> uint key (for atomic max on floats) ----
__device__ __forceinline__ unsigned f2key(float f) {
  unsigned u = __float_as_uint(f);
  return (u & 0x80000000u) ? ~u : (u | 0x80000000u);
}
__device__ __forceinline__ float key2f(unsigned k) {
  unsigned u = (k & 0x80000000u) ? (k ^ 0x80000000u) : ~k;
  return __uint_as_float(u);
}

// pad node_feats [N,6] -> [N,8] (zero pad) so every projection has K%4==0
__global__ void pad_x(const float* __restrict__ x6, float* __restrict__ x8, int N) {
  int n = blockIdx.x * blockDim.x + threadIdx.x;
  if (n >= N) return;
  const float* s = x6 + (size_t)n * 6;
  float* d = x8 + (size_t)n * 8;
#pragma unroll
  for (int i = 0; i < 6; ++i) d[i] = s[i];
  d[6] = 0.f;
  d[7] = 0.f;
}

// ------------------------------------------------------------------
// ft[N,FS] = x[N,K] @ W[HD,K]^T  via V_WMMA_F32_16X16X4_F32
// Fully unrolled K chain; unguarded aligned v2f loads (addresses clamped,
// values selected); wave-uniform tile bounds; FS-padded stores.
// A layout: lane l holds row M=l&15, K-pair (l>>4)*2.
// B layout: lane l holds col N=l&15, K-pair (l>>4)*2.
// D layout: VGPR r -> row r + 8*(l>>4), col l&15.
// ------------------------------------------------------------------
template <int K, int HD, int FS>
__global__ void proj_wmma(const float* __restrict__ x, const float* __restrict__ Wm,
                          float* __restrict__ ft, int nvalid, int nodeTiles) {
  constexpr int colTiles = FS / 16;
  constexpr int KT = K / 4;
  int wave = blockIdx.x * (blockDim.x >> 5) + (threadIdx.x >> 5);
  if (wave >= nodeTiles * colTiles) return;      // wave-uniform exit
  int nt = wave / colTiles;
  int ct = wave - nt * colTiles;
  int lane = threadIdx.x & 31;
  int half = lane >> 4;
  int mrow = lane & 15;
  int node = nt * 16 + mrow;
  int nodec = node < nvalid ? node : nvalid - 1; // clamp: garbage rows never stored
  int ncol = ct * 16 + mrow;

  v2f a[KT], b[KT];
#pragma unroll
  for (int t = 0; t < KT; ++t) {
    int ka = t * 4 + half * 2;
    a[t] = *(const v2f*)(x + (size_t)nodec * K + ka);
    if constexpr (HD < FS) {
      int ncolc = ncol < HD ? ncol : HD - 1;
      v2f bw = *(const v2f*)(Wm + (size_t)ncolc * K + ka);
      bool nok = ncol < HD;
      b[t].x = nok ? bw.x : 0.f;                 // branchless cndmask
      b[t].y = nok ? bw.y : 0.f;
    } else {
      b[t] = *(const v2f*)(Wm + (size_t)ncol * K + ka);
    }
  }
  v8f acc = {0.f, 0.f, 0.f, 0.f, 0.f, 0.f, 0.f, 0.f};
#pragma unroll
  for (int t = 0; t < KT; ++t)
    acc = __builtin_amdgcn_wmma_f32_16x16x4_f32(false, a[t], false, b[t],
                                                (short)0, acc, false, false);
  if (nt * 16 + 16 <= nvalid) {                  // uniform fast path (always, N%16==0)
#pragma unroll
    for (int r = 0; r < 8; ++r)
      ft[(size_t)(nt * 16 + r + half * 8) * FS + ncol] = acc[r];
  } else {
#pragma unroll
    for (int r = 0; r < 8; ++r) {
      int row = nt * 16 + r + half * 8;
      if (row < nvalid) ft[(size_t)row * FS + ncol] = acc[r];
    }
  }
}

// el/er per (node,head); zero m-keys, denom, acc; compute c[h]=dot(We_h,ae_h)
__global__ void node_prep(const float* __restrict__ ft,
                          const float* __restrict__ al, const float* __restrict__ ar,
                          const float* __restrict__ We, const float* __restrict__ ae,
                          float* __restrict__ el, float* __restrict__ er,
                          unsigned* __restrict__ mk, float* __restrict__ den,
                          float* __restrict__ acc, float* __restrict__ cbuf,
                          int N, int D, int HD, int FS) {
  if (blockIdx.x == 0 && threadIdx.x < 3) {
    float c = 0.f;
    for (int d = 0; d < D; ++d) c += We[threadIdx.x * D + d] * ae[threadIdx.x * D + d];
    cbuf[threadIdx.x] = c;
  }
  int n = blockIdx.x * blockDim.x + threadIdx.x;
  if (n >= N) return;
  const float* f = ft + (size_t)n * FS;
  float* a = acc + (size_t)n * FS;
#pragma unroll
  for (int h = 0; h < 3; ++h) {
    float sl = 0.f, sr = 0.f;
    for (int d = 0; d < D; ++d) {
      float v = f[h * D + d];
      sl += v * al[h * D + d];
      sr += v * ar[h * D + d];
    }
    el[n * 3 + h] = sl;
    er[n * 3 + h] = sr;
    mk[n * 3 + h] = 0u;   // below key(-inf); only ever read if the node has edges
    den[n * 3 + h] = 0.f;
  }
  for (int j = 0; j < HD; ++j) a[j] = 0.f;
}

// e = leaky0.2(el[src]+er[dst]+ef*c[h]); store; segment-max via key atomicMax
__global__ void edge_logits(const int* __restrict__ src, const int* __restrict__ dst,
                            const float* __restrict__ ef,
                            const float* __restrict__ el, const float* __restrict__ er,
                            const float* __restrict__ cbuf,
                            float* __restrict__ ews, unsigned* __restrict__ mk, int E) {
  int e = blockIdx.x * blockDim.x + threadIdx.x;
  if (e >= E) return;
  int s = src[e], t = dst[e];
  float f = ef[e];
#pragma unroll
  for (int h = 0; h < 3; ++h) {
    float v = el[s * 3 + h] + er[t * 3 + h] + f * cbuf[h];
    v = (v > 0.f) ? v : 0.2f * v;
    ews[(size_t)e * 3 + h] = v;
    atomicMax(&mk[t * 3 + h], f2key(v));
  }
}

// ex = exp(e - m[dst]); denom[dst]+=ex; acc[dst,h,:] += ex * ft[src,h,:]
// (normalization deferred to the node pass -> one less edge pass)
__global__ void edge_accum(const int* __restrict__ src, const int* __restrict__ dst,
                           const float* __restrict__ ews, const unsigned* __restrict__ mk,
                           float* __restrict__ den, const float* __restrict__ ft,
                           float* __restrict__ acc, int E, int D, int FS) {
  int e = blockIdx.x * blockDim.x + threadIdx.x;
  int h = blockIdx.y;
  if (e >= E) return;
  int s = src[e], t = dst[e];
  float v = ews[(size_t)e * 3 + h];
  float m = key2f(mk[t * 3 + h]);
  float ex = __expf(v - m);
  atomicAdd(&den[t * 3 + h], ex);
  const float4* f4 = reinterpret_cast<const float4*>(ft + (size_t)s * FS + h * D);
  float* ap = acc + (size_t)t * FS + h * D;
  for (int d4 = 0; d4 < (D >> 2); ++d4) {
    float4 fv = f4[d4];
    atomicAdd(&ap[d4 * 4 + 0], ex * fv.x);
    atomicAdd(&ap[d4 * 4 + 1], ex * fv.y);
    atomicAdd(&ap[d4 * 4 + 2], ex * fv.z);
    atomicAdd(&ap[d4 * 4 + 3], ex * fv.w);
  }
}

// layers 1-2: out = acc/denom + b, then mean over heads -> next layer input
__global__ void node_finish_mean(const float* __restrict__ acc, const float* __restrict__ den,
                                 const float* __restrict__ bias, float* __restrict__ xout,
                                 int N, int D, int FS) {
  int n = blockIdx.x * blockDim.x + threadIdx.x;
  if (n >= N) return;
  const float* a = acc + (size_t)n * FS;
  float dn[3];
#pragma unroll
  for (int h = 0; h < 3; ++h) {
    float d = den[n * 3 + h];
    dn[h] = d > 1e-9f ? d : 1e-9f;
  }
  for (int d = 0; d < D; ++d) {
    float s = 0.f;
#pragma unroll
    for (int h = 0; h < 3; ++h) s += a[h * D + d] / dn[h] + bias[h * D + d];
    xout[(size_t)n * D + d] = s * (1.f / 3.f);
  }
}

// layer 3: out = acc/denom + b, fused per-head MLP 32->16->4->1, l4 combine,
// block max-reduce, global atomic max (key trick)
__global__ void node_finish_mlp(const float* __restrict__ acc, const float* __restrict__ den,
                                const float* __restrict__ bias,
                                const float* __restrict__ l1w, const float* __restrict__ l1b,
                                const float* __restrict__ l2w, const float* __restrict__ l2b,
                                const float* __restrict__ l3w, const float* __restrict__ l3b,
                                const float* __restrict__ l4w, const float* __restrict__ l4b,
                                unsigned* __restrict__ outkey, int N) {
  __shared__ float sw1[16 * 32], sb1[16], sw2[4 * 16], sb2[4], sw3[4], sw4[3];
  __shared__ float sb3, sb4;
  __shared__ float red[256];
  for (int i = threadIdx.x; i < 16 * 32; i += blockDim.x) sw1[i] = l1w[i];
  for (int i = threadIdx.x; i < 16; i += blockDim.x) sb1[i] = l1b[i];
  for (int i = threadIdx.x; i < 64; i += blockDim.x) sw2[i] = l2w[i];
  if (threadIdx.x < 4) { sb2[threadIdx.x] = l2b[threadIdx.x]; sw3[threadIdx.x] = l3w[threadIdx.x]; }
  if (threadIdx.x < 3) sw4[threadIdx.x] = l4w[threadIdx.x];
  if (threadIdx.x == 0) { sb3 = l3b[0]; sb4 = l4b[0]; }
  __syncthreads();

  int n = blockIdx.x * blockDim.x + threadIdx.x;
  float feat = -INFINITY;
  if (n < N) {
    float hv[3];
#pragma unroll
    for (int h = 0; h < 3; ++h) {
      float d = den[n * 3 + h];
      float dn = d > 1e-9f ? d : 1e-9f;
      const float* a = acc + (size_t)n * 96 + h * 32;
      float x[32];
#pragma unroll
      for (int i = 0; i < 32; ++i) x[i] = a[i] / dn + bias[h * 32 + i];
      float y1[16];
#pragma unroll
      for (int j = 0; j < 16; ++j) {
        float s = sb1[j];
#pragma unroll
        for (int i = 0; i < 32; ++i) s += x[i] * sw1[j * 32 + i];
        y1[j] = s > 0.f ? s : 0.01f * s;
      }
      float y2[4];
#pragma unroll
      for (int j = 0; j < 4; ++j) {
        float s = sb2[j];
#pragma unroll
        for (int i = 0; i < 16; ++i) s += y1[i] * sw2[j * 16 + i];
        y2[j] = s > 0.f ? s : 0.01f * s;
      }
      float s3 = sb3;
#pragma unroll
      for (int i = 0; i < 4; ++i) s3 += y2[i] * sw3[i];
      hv[h] = s3;
    }
    feat = sb4;
#pragma unroll
    for (int h = 0; h < 3; ++h) feat += hv[h] * sw4[h];
  }
  red[threadIdx.x] = feat;
  __syncthreads();
  for (int s = blockDim.x / 2; s > 0; s >>= 1) {
    if (threadIdx.x < s) red[threadIdx.x] = fmaxf(red[threadIdx.x], red[threadIdx.x + s]);
    __syncthreads();
  }
  if (threadIdx.x == 0) atomicMax(outkey, f2key(red[0]));
}

__global__ void init_key(unsigned* __restrict__ k) { *k = 0u; }
__global__ void finalize(const unsigned* __restrict__ k, float* __restrict__ out) {
  *out = key2f(*k);
}

extern "C" void kernel_launch(void* const* d_in, const int* in_sizes, int n_in,
                              void* d_out, int out_size, void* d_ws, size_t ws_size,
                              hipStream_t stream) {
  const float* node_feats = (const float*)d_in[0];
  const float* edge_feats = (const float*)d_in[1];
  const int* src = (const int*)d_in[2];
  const int* dst = (const int*)d_in[3];
  const int N = in_sizes[0] / 6;
  const int E = in_sizes[2];

  const float* W[3]  = {(const float*)d_in[4],  (const float*)d_in[10], (const float*)d_in[16]};
  const float* We[3] = {(const float*)d_in[5],  (const float*)d_in[11], (const float*)d_in[17]};
  const float* al[3] = {(const float*)d_in[6],  (const float*)d_in[12], (const float*)d_in[18]};
  const float* ar[3] = {(const float*)d_in[7],  (const float*)d_in[13], (const float*)d_in[19]};
  const float* ae[3] = {(const float*)d_in[8],  (const float*)d_in[14], (const float*)d_in[20]};
  const float* bb[3] = {(const float*)d_in[9],  (const float*)d_in[15], (const float*)d_in[21]};
  const float* l1w = (const float*)d_in[22]; const float* l1b = (const float*)d_in[23];
  const float* l2w = (const float*)d_in[24]; const float* l2b = (const float*)d_in[25];
  const float* l3w = (const float*)d_in[26]; const float* l3b = (const float*)d_in[27];
  const float* l4w = (const float*)d_in[28]; const float* l4b = (const float*)d_in[29];

  // workspace layout (~102 MB for N=100k, E=1.6M)
  float* ws = (float*)d_ws;
  size_t off = 0;
  float* ft  = ws + off; off += (size_t)N * 96;
  float* acc = ws + off; off += (size_t)N * 96;
  float* el  = ws + off; off += (size_t)N * 3;
  float* er  = ws + off; off += (size_t)N * 3;
  unsigned* mk = (unsigned*)(ws + off); off += (size_t)N * 3;
  float* den = ws + off; off += (size_t)N * 3;
  float* ews = ws + off; off += (size_t)E * 3;
  float* xin = ws + off; off += (size_t)N * 16;
  float* cbuf = ws + off; off += 4;
  unsigned* outkey = (unsigned*)(ws + off);

  const int Dout[3] = {8, 16, 32};
  const int HD[3]   = {24, 48, 96};
  const int FS[3]   = {32, 48, 96};   // padded feature row stride (16-multiple)

  init_key<<<1, 1, 0, stream>>>(outkey);

  int nodeBlocks = (N + 255) / 256;
  int edgeBlocks = (E + 255) / 256;
  int nodeTiles = (N + 15) / 16;

  // zero-pad the 6-wide input to 8 so layer-0 WMMA K-loop is unguarded
  pad_x<<<nodeBlocks, 256, 0, stream>>>(node_feats, xin, N);

  for (int l = 0; l < 3; ++l) {
    int colTiles = FS[l] / 16;
    int projBlocks = (nodeTiles * colTiles + 7) / 8;  // 256 threads = 8 waves/block
    if (l == 0)
      proj_wmma<8, 24, 32><<<projBlocks, 256, 0, stream>>>(xin, W[0], ft, N, nodeTiles);
    else if (l == 1)
      proj_wmma<8, 48, 48><<<projBlocks, 256, 0, stream>>>(xin, W[1], ft, N, nodeTiles);
    else
      proj_wmma<16, 96, 96><<<projBlocks, 256, 0, stream>>>(xin, W[2], ft, N, nodeTiles);

    node_prep<<<nodeBlocks, 256, 0, stream>>>(ft, al[l], ar[l], We[l], ae[l],
                                              el, er, mk, den, acc, cbuf,
                                              N, Dout[l], HD[l], FS[l]);
    edge_logits<<<edgeBlocks, 256, 0, stream>>>(src, dst, edge_feats, el, er,
                                                cbuf, ews, mk, E);
    edge_accum<<<dim3(edgeBlocks, 3), 256, 0, stream>>>(src, dst, ews, mk, den,
                                                        ft, acc, E, Dout[l], FS[l]);
    if (l < 2) {
      node_finish_mean<<<nodeBlocks, 256, 0, stream>>>(acc, den, bb[l], xin,
                                                       N, Dout[l], FS[l]);
    } else {
      node_finish_mlp<<<nodeBlocks, 256, 0, stream>>>(acc, den, bb[l],
                                                      l1w, l1b, l2w, l2b,
                                                      l3w, l3b, l4w, l4b,
                                                      outkey, N);
    }
  }
  finalize<<<1, 1, 0, stream>>>(outkey, (float*)d_out);
}